// FFFTrainFixed_55078660604599
// MI455X (gfx1250) — compile-verified
//
#include <hip/hip_runtime.h>
#include <hip/hip_bf16.h>
#include <math.h>

// ---------------------------------------------------------------------------
// FFF (fast feed-forward) tree inference for MI455X / gfx1250 (wave32).
//
//   oldx  : [8192, 768]  f32
//   w_in  : [4095, 768]  f32   (node -> input weights, row-major: rows gatherable)
//   w_out : [768, 4095]  f32   (transposed once into ws so node columns are rows)
//   out   : [8192, 768]  f32
//
// Pipeline (all on `stream`, serialized):
//   K0 transpose_wout : w_out -> w_outT [4095,768] in d_ws
//   K1 fff_walk       : wave-per-token tree walk; writes 12 (node, gelu) pairs
//                       per token into d_ws. Uses global_prefetch_b8 on both
//                       children to hide the dependent-gather latency chain,
//                       and wave32 shfl_xor butterfly reductions.
//   K2 fff_out_wmma   : per 16-token x 128-col tile, accumulates
//                       out += diag(g_level) * w_outT[nodes_level] over the 12
//                       levels with V_WMMA_F32_16X16X4_F32 (exact f32 math).
// ---------------------------------------------------------------------------

#define N_TOKENS   8192
#define IN_W       768
#define OUT_W      768
#define N_NODES    4095
#define DEPTH      11
#define N_LEVELS   (DEPTH + 1)      // 12 active nodes per token
#define NCG        6                // column groups of 128 in K2 (768/128)

typedef __attribute__((ext_vector_type(2))) float v2f;
typedef __attribute__((ext_vector_type(8))) float v8f;

// ---------------- K0: w_out [768 x 4095] -> w_outT [4095 x 768] -------------
__global__ void transpose_wout(const float* __restrict__ w_out,
                               float* __restrict__ wT) {
  __shared__ float t[32][33];
  const int tx = threadIdx.x;          // 0..31
  const int ty = threadIdx.y;          // 0..7
  const int n0 = blockIdx.x * 32;      // node tile (4095 -> 128 tiles)
  const int c0 = blockIdx.y * 32;      // out-width tile (768 -> 24 tiles)
  const int n  = n0 + tx;
#pragma unroll
  for (int i = 0; i < 32; i += 8) {
    const int c = c0 + ty + i;
    t[ty + i][tx] = (n < N_NODES) ? w_out[(size_t)c * N_NODES + n] : 0.0f;
  }
  __syncthreads();
#pragma unroll
  for (int i = 0; i < 32; i += 8) {
    const int row = n0 + ty + i;       // wT row = node
    if (row < N_NODES) wT[(size_t)row * OUT_W + (c0 + tx)] = t[tx][ty + i];
  }
}

// ---------------- K1: wave-per-token tree walk ------------------------------
__global__ void fff_walk(const float* __restrict__ x,
                         const float* __restrict__ w_in,
                         float* __restrict__ acts,    // [N_TOKENS][12]
                         int*   __restrict__ nodes) { // [N_TOKENS][12]
  const int lane  = threadIdx.x & 31;
  const int token = blockIdx.x * (blockDim.x >> 5) + (threadIdx.x >> 5);

  // x row resident in registers: 6 x float4 per lane = 768 floats per wave.
  const float4* xr = (const float4*)(x + (size_t)token * IN_W);
  float4 xv[6];
#pragma unroll
  for (int j = 0; j < 6; ++j) xv[j] = xr[lane + 32 * j];

  int n = 0;                                    // root
  for (int d = 0; d < N_LEVELS; ++d) {
    // Prefetch both children's w_in rows (3 KB each; 32 lanes * 128B lines
    // cover a row) while we reduce the current node -> hides the 12-deep
    // dependent gather chain; extra traffic is L2-resident w_in.
    if (d < DEPTH) {
      __builtin_prefetch(w_in + (size_t)(2 * n + 1) * IN_W + lane * 32, 0, 3);
      __builtin_prefetch(w_in + (size_t)(2 * n + 2) * IN_W + lane * 32, 0, 3);
    }
    const float4* wr = (const float4*)(w_in + (size_t)n * IN_W);
    float acc = 0.0f;
#pragma unroll
    for (int j = 0; j < 6; ++j) {
      const float4 w = wr[lane + 32 * j];
      acc = fmaf(xv[j].x, w.x, acc);
      acc = fmaf(xv[j].y, w.y, acc);
      acc = fmaf(xv[j].z, w.z, acc);
      acc = fmaf(xv[j].w, w.w, acc);
    }
    // wave32 butterfly reduction -> every lane holds the full dot product
#pragma unroll
    for (int off = 16; off > 0; off >>= 1) acc += __shfl_xor(acc, off, 32);

    // exact (erf) GELU, matching torch nn.GELU default / jax approximate=False
    const float act = 0.5f * acc * (1.0f + erff(acc * 0.70710678118654752f));
    if (lane == 0) {
      acts[token * N_LEVELS + d]  = act;
      nodes[token * N_LEVELS + d] = n;
    }
    n = 2 * n + 1 + (acc > 0.0f ? 1 : 0);       // decision: strictly > 0
  }
}

// ---------------- K2: out += diag(g) * gathered w_outT rows via WMMA --------
// One wave handles a 16-token x 128-column tile of out.
// Per level d: out[i, :] += g_i * w_outT[n_i, :]. Expressed as 16x16x16 f32
// matmul with diagonal A, split into four K=4 V_WMMA_F32_16X16X4_F32 steps.
// Exact f32 multiply/accumulate -> numerically equivalent to the VALU version.
__global__ void fff_out_wmma(const float* __restrict__ wT,
                             const float* __restrict__ acts,
                             const int*   __restrict__ nodes,
                             float* __restrict__ out) {
  const int lane = threadIdx.x & 31;
  const int wave = blockIdx.x * (blockDim.x >> 5) + (threadIdx.x >> 5);
  const int tile = wave / NCG;          // 16-token tile index (0..511)
  const int cg   = wave % NCG;          // 128-col group (0..5)
  const int tok0 = tile * 16;
  const int c0   = cg * 128;
  const int half = lane >> 4;           // 0: lanes 0-15, 1: lanes 16-31
  const int l15  = lane & 15;

  v8f C[8];                             // 8 chunks of 16 cols: 16x128 f32 tile
#pragma unroll
  for (int j = 0; j < 8; ++j) C[j] = v8f{0.f, 0.f, 0.f, 0.f, 0.f, 0.f, 0.f, 0.f};

  for (int d = 0; d < N_LEVELS; ++d) {
    // lane i (and i+16) hold token (tok0+i)'s gelu value + node for level d
    const float g = acts[(size_t)(tok0 + l15) * N_LEVELS + d];
    const int   n = nodes[(size_t)(tok0 + l15) * N_LEVELS + d];
#pragma unroll
    for (int s = 0; s < 4; ++s) {
      // A (16x4, ISA layout: lanes0-15 = K0/K1, lanes16-31 = K2/K3):
      // A[i][k] = g_i iff i == 4s+k  (diagonal slice for tokens 4s..4s+3)
      v2f A;
      A.x = (lane == 4 * s)          ? g : ((lane == 16 + 4 * s + 2) ? g : 0.0f);
      A.y = (lane == 4 * s + 1)      ? g : ((lane == 16 + 4 * s + 3) ? g : 0.0f);
      // B (4x16): row k = w_outT[node of token 4s+k].  ISA layout:
      // VGPR0 = rows {0,2}, VGPR1 = rows {1,3}, halves of the wave.
      const int r02 = __shfl(n, 4 * s + 2 * half, 32);      // rows 0 / 2
      const int r13 = __shfl(n, 4 * s + 1 + 2 * half, 32);  // rows 1 / 3
#pragma unroll
      for (int j = 0; j < 8; ++j) {
        v2f B;
        B.x = wT[(size_t)r02 * OUT_W + c0 + 16 * j + l15];
        B.y = wT[(size_t)r13 * OUT_W + c0 + 16 * j + l15];
        // (neg_a, A, neg_b, B, c_mod, C, reuse_a, reuse_b)
        C[j] = __builtin_amdgcn_wmma_f32_16x16x4_f32(
            false, A, false, B, (short)0, C[j], false, false);
      }
    }
  }
  // C layout: VGPR r -> rows {r, r+8}, N = lane within half-wave.
#pragma unroll
  for (int j = 0; j < 8; ++j) {
#pragma unroll
    for (int r = 0; r < 8; ++r) {
      out[(size_t)(tok0 + r + 8 * half) * OUT_W + c0 + 16 * j + l15] = C[j][r];
    }
  }
}

// ---------------------------------------------------------------------------
extern "C" void kernel_launch(void* const* d_in, const int* in_sizes, int n_in,
                              void* d_out, int out_size, void* d_ws, size_t ws_size,
                              hipStream_t stream) {
  const float* oldx  = (const float*)d_in[0];
  const float* w_in  = (const float*)d_in[1];
  const float* w_out = (const float*)d_in[2];
  float* out = (float*)d_out;

  // workspace layout (needs ~13.37 MB)
  const size_t WT_BYTES  = (size_t)N_NODES * OUT_W * sizeof(float);   // 12,579,840
  const size_t ACT_BYTES = (size_t)N_TOKENS * N_LEVELS * sizeof(float); // 393,216
  char* ws = (char*)d_ws;
  float* wT    = (float*)ws;
  float* acts  = (float*)(ws + WT_BYTES);
  int*   nodeb = (int*)(ws + WT_BYTES + ACT_BYTES);

  // K0: transpose w_out so gathered node columns are contiguous rows.
  transpose_wout<<<dim3(128, 24), dim3(32, 8), 0, stream>>>(w_out, wT);

  // K1: 8192 tokens, one wave32 each; 256 threads = 8 waves per block.
  fff_walk<<<N_TOKENS / 8, 256, 0, stream>>>(oldx, w_in, acts, nodeb);

  // K2: 512 token-tiles * 6 col-groups = 3072 waves; 4 waves per block.
  fff_out_wmma<<<(512 * NCG) / 4, 128, 0, stream>>>(wT, acts, nodeb, out);
}